// DetectionLayer_51625506898668
// MI455X (gfx1250) — compile-verified
//
#include <hip/hip_runtime.h>
#include <cstdint>

#define BS 32
#define NA 9
#define FH 100
#define FW 168
#define FHW (FH*FW)            // 16800
#define NBOX (NA*FHW)          // 151200
#define TPB 1024               // 32 waves (wave32)
#define CHUNK 148              // ceil(NBOX/TPB)
#define NP (CHUNK*TPB)         // 151552 padded
#define KEEP_N 300
#define NEGV (-1e30f)
#define IOU_THR 0.7f
#define CONF_THR 0.5f

// ---------------- decode: softmax score + anchor box decode (memory bound) ---------------
__global__ __launch_bounds__(256) void decode_kernel(
    const float* __restrict__ preds, const float* __restrict__ regs,
    const int* __restrict__ ih, const int* __restrict__ iw,
    float4* __restrict__ boxes, float* __restrict__ scores,
    int* __restrict__ keep, float* __restrict__ keepScore)
{
  long long tid = (long long)blockIdx.x * blockDim.x + threadIdx.x;
  if (tid < BS * KEEP_N) { keep[tid] = -1; keepScore[tid] = NEGV; }
  if (tid >= (long long)BS * NP) return;
  int b = (int)(tid / NP);
  int n = (int)(tid % NP);
  if (n >= NBOX) {  // padding: never selectable
    scores[tid] = NEGV;
    boxes[tid] = make_float4(0.f, 0.f, 0.f, 0.f);
    return;
  }
  int a   = n / FHW;
  int rem = n - a * FHW;
  int y   = rem / FW;
  int x   = rem - y * FW;

  const float RT[3] = {0.5f, 1.0f, 2.0f};
  const float SC[3] = {0.5f, 1.0f, 2.0f};
  double r = (double)RT[a / 3], sA = (double)SC[a % 3];
  double sq = sqrt(r);
  float wa = (float)(16.0 * sA / sq);   // STRIDE*s*sqrt(1/r)
  float ha = (float)(16.0 * sA * sq);   // STRIDE*s*sqrt(r)

  size_t pb = (size_t)b * (2 * NA * FHW);
  float p0 = preds[pb + (size_t)a * FHW + rem];
  float p1 = preds[pb + (size_t)(NA + a) * FHW + rem];
  float sc = 1.0f / (1.0f + expf(p0 - p1));   // softmax[...,1]

  size_t rb = (size_t)b * (4 * NA * FHW);
  float d0 = regs[rb + (size_t)(0 * NA + a) * FHW + rem];
  float d1 = regs[rb + (size_t)(1 * NA + a) * FHW + rem];
  float d2 = regs[rb + (size_t)(2 * NA + a) * FHW + rem];
  float d3 = regs[rb + (size_t)(3 * NA + a) * FHW + rem];

  float W = (float)iw[0], H = (float)ih[0];
  float cx = d0 * wa + (float)x * 16.0f;
  float cy = d1 * ha + (float)y * 16.0f;
  float w  = wa * expf(d2);
  float h  = ha * expf(d3);
  float x1 = fminf(fmaxf(cx - 0.5f * w, 0.0f), W);
  float y1 = fminf(fmaxf(cy - 0.5f * h, 0.0f), H);
  float x2 = fminf(fmaxf(cx + 0.5f * w, 0.0f), W);
  float y2 = fminf(fmaxf(cy + 0.5f * h, 0.0f), H);

  scores[tid] = sc;
  boxes[tid] = make_float4(x1, y1, x2, y2);
}

// ---------------- wave32 argmax reduce (first-occurrence tie-break) ---------------
__device__ __forceinline__ void wave_argmax(float &v, int &idx) {
#pragma unroll
  for (int off = 16; off > 0; off >>= 1) {
    float ov = __shfl_down(v, off, 32);
    int   oi = __shfl_down(idx, off, 32);
    if (ov > v || (ov == v && oi < idx)) { v = ov; idx = oi; }
  }
}

// ---------------- greedy NMS: 1 block per image; fused suppress + next-argmax ---------------
__global__ __launch_bounds__(TPB) void nms_kernel(
    const float4* __restrict__ boxesAll, float* __restrict__ scoresAll,
    int* __restrict__ keep, float* __restrict__ keepScore)
{
  __shared__ float4 stage[2][TPB];   // double-buffered async-staged box tiles
  __shared__ float  redV[32];
  __shared__ int    redI[32];
  __shared__ float  sBV;
  __shared__ int    sBI;

  const int b = blockIdx.x;
  const int t = threadIdx.x;
  const int lane = t & 31, wv = t >> 5;
  const float4* __restrict__ bx = boxesAll + (size_t)b * NP;
  float* __restrict__ s = scoresAll + (size_t)b * NP;

  const uint32_t stageBase = (uint32_t)(uintptr_t)(void*)&stage[0][0];
  const uint32_t slotOff   = (uint32_t)t * 16u;

  // initial argmax over scores (no boxes needed)
  float bv = -3.0e38f; int bi = 0;
  for (int c = 0; c < CHUNK; ++c) {
    int i = c * TPB + t;
    float v = s[i];
    if (v > bv) { bv = v; bi = i; }
  }
  wave_argmax(bv, bi);
  if (lane == 0) { redV[wv] = bv; redI[wv] = bi; }
  __syncthreads();
  if (wv == 0) {
    bv = redV[lane]; bi = redI[lane];
    wave_argmax(bv, bi);
    if (lane == 0) { sBV = bv; sBI = bi; }
  }
  __syncthreads();

  for (int it = 0; it < KEEP_N; ++it) {
    float selV = sBV;
    int   selI = sBI;
    if (selV <= NEGV * 0.5f) break;          // uniform: remaining keep stay -1
    if (t == 0) {
      keep[b * KEEP_N + it]      = selI;
      keepScore[b * KEEP_N + it] = selV;     // == original score at selection time
    }
    float4 bj = bx[selI];
    float areaJ = fmaxf(bj.z - bj.x, 0.0f) * fmaxf(bj.w - bj.y, 0.0f);

    bv = -3.0e38f; bi = 0;
    // prologue: async-stage chunk 0 into buffer 0 (CDNA5 async copy engine)
    asm volatile("s_wait_dscnt 0x0" ::: "memory");
    asm volatile("global_load_async_to_lds_b128 %0, %1, off"
                 :: "v"(stageBase + slotOff), "v"(bx + t) : "memory");
    for (int c = 0; c < CHUNK; ++c) {
      const int i = c * TPB + t;
      if (c + 1 < CHUNK) {
        uint32_t dst = stageBase + (uint32_t)(((c + 1) & 1) * TPB) * 16u + slotOff;
        asm volatile("s_wait_dscnt 0x0" ::: "memory");   // prior buffer read drained
        asm volatile("global_load_async_to_lds_b128 %0, %1, off"
                     :: "v"(dst), "v"(bx + i + TPB) : "memory");
        asm volatile("s_wait_asynccnt 0x1" ::: "memory"); // chunk c landed; c+1 in flight
      } else {
        asm volatile("s_wait_asynccnt 0x0" ::: "memory");
      }
      float4 B = stage[c & 1][t];   // each lane consumes only its own wave's async data
      float v = s[i];
      if (v > NEGV * 0.5f) {        // dead entries stay NEG (value-equivalent to reference)
        float ix1 = fmaxf(B.x, bj.x);
        float iy1 = fmaxf(B.y, bj.y);
        float ix2 = fminf(B.z, bj.z);
        float iy2 = fminf(B.w, bj.w);
        float inter = fmaxf(ix2 - ix1, 0.0f) * fmaxf(iy2 - iy1, 0.0f);
        float area  = fmaxf(B.z - B.x, 0.0f) * fmaxf(B.w - B.y, 0.0f);
        float iou   = inter / (area + areaJ - inter + 1e-9f);
        if (iou > IOU_THR || i == selI) { v = NEGV; s[i] = NEGV; }
      }
      if (v > bv) { bv = v; bi = i; }        // fused next-iteration argmax
    }
    wave_argmax(bv, bi);
    if (lane == 0) { redV[wv] = bv; redI[wv] = bi; }
    __syncthreads();
    if (wv == 0) {
      bv = redV[lane]; bi = redI[lane];
      wave_argmax(bv, bi);
      if (lane == 0) { sBV = bv; sBI = bi; }
    }
    __syncthreads();
  }
}

// ---------------- gather detections ---------------
__global__ __launch_bounds__(256) void out_kernel(
    const float4* __restrict__ boxesAll, const int* __restrict__ keep,
    const float* __restrict__ keepScore, float* __restrict__ out)
{
  int t = blockIdx.x * blockDim.x + threadIdx.x;
  if (t >= BS * KEEP_N) return;
  int b = t / KEEP_N;
  int j = keep[t];
  float sc = keepScore[t];
  bool valid = (j >= 0) && (sc > CONF_THR);
  float4 bo = make_float4(0.f, 0.f, 0.f, 0.f);
  if (valid) bo = boxesAll[(size_t)b * NP + j];
  float* o = out + (size_t)t * 5;
  o[0] = valid ? bo.x : 0.0f;
  o[1] = valid ? bo.y : 0.0f;
  o[2] = valid ? bo.z : 0.0f;
  o[3] = valid ? bo.w : 0.0f;
  o[4] = valid ? sc   : 0.0f;
}

extern "C" void kernel_launch(void* const* d_in, const int* in_sizes, int n_in,
                              void* d_out, int out_size, void* d_ws, size_t ws_size,
                              hipStream_t stream) {
  const float* preds = (const float*)d_in[0];
  const float* regs  = (const float*)d_in[1];
  const int*   ih    = (const int*)d_in[2];
  const int*   iw    = (const int*)d_in[3];

  uint8_t* ws = (uint8_t*)d_ws;
  size_t boxesBytes  = (size_t)BS * NP * sizeof(float4);   // ~77.6 MB
  size_t scoresBytes = (size_t)BS * NP * sizeof(float);    // ~19.4 MB
  size_t keepBytes   = (size_t)BS * KEEP_N * sizeof(int);
  float4* boxes     = (float4*)ws;
  float*  scores    = (float*)(ws + boxesBytes);
  int*    keep      = (int*)(ws + boxesBytes + scoresBytes);
  float*  keepScore = (float*)(ws + boxesBytes + scoresBytes + keepBytes);

  long long total = (long long)BS * NP;
  int blocks = (int)((total + 255) / 256);
  hipLaunchKernelGGL(decode_kernel, dim3(blocks), dim3(256), 0, stream,
                     preds, regs, ih, iw, boxes, scores, keep, keepScore);
  hipLaunchKernelGGL(nms_kernel, dim3(BS), dim3(TPB), 0, stream,
                     boxes, scores, keep, keepScore);
  hipLaunchKernelGGL(out_kernel, dim3((BS * KEEP_N + 255) / 256), dim3(256), 0, stream,
                     boxes, keep, keepScore, (float*)d_out);
}